// TypedBinaryTreeLSTM_54219667145454
// MI455X (gfx1250) — compile-verified
//
#include <hip/hip_runtime.h>
#include <cstdint>

#define B_    256
#define N_    16
#define M_    128
#define V_    512
#define K_    8
#define NP1   17
#define BLOCK 512
#define NW    16           // waves per block (wave32)
#define RPW   8            // rows per wave = M_/NW
#define NEGBIG -1e9f

typedef int   v4i __attribute__((vector_size(16)));
typedef float v4f __attribute__((vector_size(16)));
typedef __attribute__((address_space(1))) v4i* gptr_v4i;   // global (device) int4*
typedef __attribute__((address_space(3))) v4i* lptr_v4i;   // LDS int4*

// ---- async global->LDS b128, ASYNCcnt-tracked (CDNA5 path) ----
#if __has_builtin(__builtin_amdgcn_global_load_async_to_lds_b128)
#define ASYNC_B128(GP, LDSOFF, IMM)                                         \
  __builtin_amdgcn_global_load_async_to_lds_b128(                           \
      (gptr_v4i)(uintptr_t)(const void*)(GP),                               \
      (lptr_v4i)(uintptr_t)(uint32_t)(LDSOFF), (IMM), 0)
#else
#define ASYNC_B128(GP, LDSOFF, IMM)                                         \
  asm volatile("global_load_async_to_lds_b128 %0, %1, off offset:%c2"       \
               :: "v"((uint32_t)(LDSOFF)),                                  \
                  "v"((uint64_t)(uintptr_t)(const void*)(GP)),              \
                  "n"(IMM)                                                  \
               : "memory")
#endif

#if __has_builtin(__builtin_amdgcn_s_wait_asynccnt)
#define WAIT_ASYNC(NN) do { __builtin_amdgcn_s_wait_asynccnt(NN);           \
                            asm volatile("" ::: "memory"); } while (0)
#else
#define WAIT_ASYNC(NN) asm volatile("s_wait_asynccnt %c0" :: "n"(NN) : "memory")
#endif

// issue one 2KB row (512 floats) into an LDS slot: 4 x b128 per lane
#define ISSUE_ROW(GROW, LDSBASE, LANE) do {                                 \
    const char* _g = (const char*)(GROW) + (size_t)(LANE) * 16;             \
    uint32_t    _l = (uint32_t)(LDSBASE) + (uint32_t)(LANE) * 16;           \
    ASYNC_B128(_g, _l, 0);                                                  \
    ASYNC_B128(_g, _l, 512);                                                \
    ASYNC_B128(_g, _l, 1024);                                               \
    ASYNC_B128(_g, _l, 1536);                                               \
  } while (0)

__global__ __launch_bounds__(BLOCK)
void template_place_kernel(const float* __restrict__ decodings,   // [B,N,M,V]
                           const float* __restrict__ dec_logits,  // [B,K,N+1]
                           const float* __restrict__ noise,       // [B,K,N+1]
                           const int*   __restrict__ target_types,// [B]
                           const int*   __restrict__ spans,       // [B]
                           float*       __restrict__ out)         // [B,M,V]
{
  extern __shared__ float rowbufs[];        // NW*2*V_ floats = 64 KB (dynamic)
  __shared__ int s_flags[M_];
  __shared__ int s_nsel[K_];
  __shared__ int s_olen;

  const int b    = blockIdx.x;
  const int tid  = threadIdx.x;
  const int w    = tid >> 5;
  const int lane = tid & 31;

  // ---- phase 0: per-(b,k) hard gumbel argmax over 17 masked logits ----
  // forward value of straight-through gumbel-softmax(hard) is exactly the
  // one-hot of argmax(log_softmax(masked) + gumbel); the log-softmax shift is
  // uniform per row so argmax(masked + gumbel) is identical.
  if (tid < K_) {
    int nsel;
    if (target_types[b] == 20) {
      nsel = (tid == 0) ? 1 : 0;            // start_template rows
    } else {
      const int span = spans[b];
      const float* lg = dec_logits + ((size_t)b * K_ + tid) * NP1;
      const float* un = noise      + ((size_t)b * K_ + tid) * NP1;
      float best = -3.4e38f; int besti = 0;
      for (int n = 0; n < NP1; ++n) {
        float val = (n <= span) ? lg[n] : NEGBIG;
        float g   = -logf(-logf(un[n]));            // gumbel noise
        float s   = val + g;
        if (s > best) { best = s; besti = n; }      // first-index tie-break
      }
      nsel = besti;
    }
    s_nsel[tid] = nsel;
  }
  __syncthreads();

  int idx = 0;                               // running placement row (uniform)
  for (int k = 0; k < K_; ++k) {
    if (idx >= M_) break;                    // remaining k are no-ops
    const int n = s_nsel[k];
    if (n == 0) continue;                    // pad tile: olen==0, nothing placed
    const float* tile = decodings + (((size_t)b * N_) + (n - 1)) * (size_t)(M_ * V_);

    if (tid == 0) s_olen = 0;
    __syncthreads();

    // ---- phase 1: per-row flag = (max(row[1:]) > row[0]) via async LDS stream ----
    {
      const uint32_t slot0 = (uint32_t)(uintptr_t)(&rowbufs[(w * 2 + 0) * V_]);
      const uint32_t slot1 = (uint32_t)(uintptr_t)(&rowbufs[(w * 2 + 1) * V_]);
      ISSUE_ROW(tile + (size_t)w * V_, slot0, lane);   // prime buffer 0 with row w
      for (int i = 0; i < RPW; ++i) {
        const int r  = w + NW * i;
        const int bb = i & 1;
        if (i + 1 < RPW) {
          ISSUE_ROW(tile + (size_t)(r + NW) * V_, (bb ? slot0 : slot1), lane);
          WAIT_ASYNC(4);                     // in-order: first 4 (current row) done
        } else {
          WAIT_ASYNC(0);
        }
        const float* rb = &rowbufs[(w * 2 + bb) * V_];
        float mx = -3.4e38f;
        #pragma unroll
        for (int j = 0; j < 16; ++j) {
          const int v = lane + j * 32;
          const float x = rb[v];
          if (v != 0) mx = fmaxf(mx, x);     // exclude v==0 from the rest-max
        }
        #pragma unroll
        for (int s = 16; s > 0; s >>= 1) mx = fmaxf(mx, __shfl_xor(mx, s));
        if (lane == 0) s_flags[r] = (mx > rb[0]) ? 1 : 0;
      }
    }
    __syncthreads();

    // ---- phase 2: olen = 1 + last flagged row, clipped to remaining rows ----
    if (tid < M_) {
      const int cand = s_flags[tid] ? (tid + 1) : 0;
      atomicMax(&s_olen, cand);
    }
    __syncthreads();
    int olen = s_olen;
    if (olen > M_ - idx) olen = M_ - idx;

    // ---- phase 3: copy rows [0,olen) -> out rows [idx, idx+olen) ----
    // tile reads are L2-hot from phase 1; output stores are non-temporal
    // (written once, never re-read) so they don't evict resident tiles.
    {
      const int n4 = olen * (V_ / 4);
      const v4f* src = (const v4f*)tile;
      v4f* dst = (v4f*)(out + ((size_t)b * M_ + idx) * V_);
      for (int j = tid; j < n4; j += BLOCK)
        __builtin_nontemporal_store(src[j], dst + j);
    }
    idx += olen;
    __syncthreads();                         // protect s_olen/s_flags reuse next k
  }

  // ---- phase 4: zero the uncovered tail rows [idx, M) ----
  if (idx < M_) {
    const int n4 = (M_ - idx) * (V_ / 4);
    v4f* dst = (v4f*)(out + ((size_t)b * M_ + idx) * V_);
    const v4f z = {0.0f, 0.0f, 0.0f, 0.0f};
    for (int j = tid; j < n4; j += BLOCK)
      __builtin_nontemporal_store(z, dst + j);
  }
}

extern "C" void kernel_launch(void* const* d_in, const int* in_sizes, int n_in,
                              void* d_out, int out_size, void* d_ws, size_t ws_size,
                              hipStream_t stream) {
  (void)in_sizes; (void)n_in; (void)out_size; (void)d_ws; (void)ws_size;
  const float* decodings     = (const float*)d_in[0];
  const float* dec_logits    = (const float*)d_in[1];
  const float* uniform_noise = (const float*)d_in[2];
  const int*   target_types  = (const int*)d_in[3];
  const int*   spans         = (const int*)d_in[4];
  float*       out           = (float*)d_out;

  const size_t shmem = (size_t)NW * 2 * V_ * sizeof(float);   // 64 KB dynamic LDS
  template_place_kernel<<<dim3(B_), dim3(BLOCK), shmem, stream>>>(
      decodings, dec_logits, uniform_noise, target_types, spans, out);
}